// AutoCorrelation_84473416778210
// MI455X (gfx1250) — compile-verified
//
#include <hip/hip_runtime.h>

// ---------------------------------------------------------------------------
// AutoCorrelation (Autoformer-style) for MI455X / gfx1250, fp32.
//
// Math simplification: R.mean over (H, L) of irfft(Qf*conj(Kf)) equals
// (1/(H*L)) * sum_h (sum_t Q)*(sum_t K), and sum_t Q = (sum_t q)@Wq + L*bq.
// => no FFT, no full Q/K projections. Heavy work = two 32768x512x512 fp32
// GEMMs (WMMA f32 16x16x4) + a 41-tap circular FIR over the length axis.
// Memory-bound: ~0.6 GB traffic @ 23.3 TB/s ~ 25-30 us; math negligible.
//
// Workspace layout (floats):
//   [0      .. 4096)   sq  [B*D]
//   [4096   .. 8192)   sk  [B*D]
//   [8192   .. 8704)   mean_value [B*64]
//   [8704   .. 8768)   idx (int)  [41]
//   [8768   .. 9280)   weights    [B*41]
//   [16384  .. +B*L*D) agg        (64 MB)
// Total required ws: ~64.1 MB.  Vp lives in d_out (consumed before final GEMM).
// ---------------------------------------------------------------------------

#define BB   8
#define LL   4096
#define DD   512
#define HH   8
#define DKK  64
#define TOPK 41

typedef float v2f __attribute__((ext_vector_type(2)));
typedef float v8f __attribute__((ext_vector_type(8)));

__global__ void zero_ws(float* p, int n) {
  int i = blockIdx.x * blockDim.x + threadIdx.x;
  if (i < n) p[i] = 0.f;
}

// Column sums of q and k over the length axis -> sq,sk [B,D]
__global__ void colsum_qk(const float* __restrict__ q, const float* __restrict__ k,
                          float* __restrict__ sq, float* __restrict__ sk) {
  const int b = blockIdx.y;
  const int d = threadIdx.x;            // 512
  const int t0 = blockIdx.x * 128;      // 32 chunks of 128 rows
  const float* qb = q + ((size_t)b * LL + t0) * DD + d;
  const float* kb = k + ((size_t)b * LL + t0) * DD + d;
  float aq = 0.f, ak = 0.f;
  for (int t = 0; t < 128; ++t) {
    aq += qb[(size_t)t * DD];
    ak += kb[(size_t)t * DD];
  }
  atomicAdd(&sq[b * DD + d], aq);
  atomicAdd(&sk[b * DD + d], ak);
}

// SQ[c] = sq[b]@Wq[:,c] + L*bq[c]; SK likewise; mean_value[b,d] = mean_h SQ*SK /(H*L)
__global__ void stats_proj(const float* __restrict__ sq, const float* __restrict__ sk,
                           const float* __restrict__ Wq, const float* __restrict__ bq,
                           const float* __restrict__ Wk, const float* __restrict__ bk,
                           float* __restrict__ mean_value) {
  __shared__ float prod[DD];
  const int b = blockIdx.x;
  const int c = threadIdx.x;            // 512
  const float* sqb = sq + b * DD;
  const float* skb = sk + b * DD;
  float accq = 0.f, acck = 0.f;
  for (int j = 0; j < DD; ++j) {
    accq += sqb[j] * Wq[(size_t)j * DD + c];
    acck += skb[j] * Wk[(size_t)j * DD + c];
  }
  accq += (float)LL * bq[c];
  acck += (float)LL * bk[c];
  prod[c] = accq * acck;
  __syncthreads();
  if (c < DKK) {
    float s = 0.f;
    for (int h = 0; h < HH; ++h) s += prod[h * DKK + c];
    mean_value[b * DKK + c] = s / (float)(HH * LL);
  }
}

// mean over batch, top-41 of 64 channel indices, per-batch softmax weights
__global__ void topk_softmax(const float* __restrict__ mean_value,
                             int* __restrict__ idx_out, float* __restrict__ w_out) {
  __shared__ float mab[DKK];
  __shared__ int sel[TOPK];
  const int t = threadIdx.x;            // 64
  float s = 0.f;
  for (int b = 0; b < BB; ++b) s += mean_value[b * DKK + t];
  mab[t] = s * (1.f / BB);
  __syncthreads();
  if (t == 0) {
    bool used[DKK] = {false};
    for (int kk = 0; kk < TOPK; ++kk) {
      int bi = 0; float bv = -3.4e38f;
      for (int d = 0; d < DKK; ++d)
        if (!used[d] && mab[d] > bv) { bv = mab[d]; bi = d; }
      used[bi] = true;
      sel[kk] = bi;
      idx_out[kk] = bi;
    }
  }
  __syncthreads();
  if (t < BB) {
    const int b = t;
    float vals[TOPK];
    float mx = -3.4e38f;
    for (int i = 0; i < TOPK; ++i) {
      vals[i] = mean_value[b * DKK + sel[i]];
      mx = fmaxf(mx, vals[i]);
    }
    float sum = 0.f;
    for (int i = 0; i < TOPK; ++i) { vals[i] = __expf(vals[i] - mx); sum += vals[i]; }
    const float inv = 1.f / sum;
    for (int i = 0; i < TOPK; ++i) w_out[b * TOPK + i] = vals[i] * inv;
  }
}

// C[M,512] = A[M,512] @ Bw[512,512] + bias, fp32 via V_WMMA_F32_16X16X4_F32.
// Block = 256 threads (8 wave32 waves) -> 64x64 output tile; K staged in LDS
// in 32-wide chunks. Fragment layouts per ISA 7.12.2:
//   A (16x4 f32, 2 VGPRs): lane l holds A[l%16, (l<16?0:2)+{0,1}]
//   B (4x16):              lane l holds B[(l<16?0:2)+{0,1}, l%16]
//   C/D (16x16, 8 VGPRs):  vgpr j, lane l -> C[j + (l<16?0:8), l%16]
__global__ __launch_bounds__(256) void gemm512_wmma(const float* __restrict__ A,
                                                    const float* __restrict__ Bw,
                                                    const float* __restrict__ bias,
                                                    float* __restrict__ C) {
  __shared__ float As[64][36];   // pitch 36: float4-aligned rows, even for v2f loads
  __shared__ float Bs[32][68];   // pitch 68: float4-aligned rows
  const int tid  = threadIdx.x;
  const int wave = tid >> 5;
  const int lane = tid & 31;
  const int row0 = blockIdx.x * 64;          // M tile
  const int col0 = blockIdx.y * 64;          // N tile
  const int tr = wave & 3;                   // 16-row sub-tile
  const int tc = (wave >> 2) * 32;           // 32-col strip -> two 16x16 tiles

  v8f acc0 = {0.f, 0.f, 0.f, 0.f, 0.f, 0.f, 0.f, 0.f};
  v8f acc1 = acc0;

  const int la_r = tid >> 3, la_c = (tid & 7) * 4;    // A stage: 64x32
  const int lb_r = tid >> 4, lb_c = (tid & 15) * 4;   // B stage: 32x64

  const int m    = lane & 15;
  const int koff = (lane < 16) ? 0 : 2;

  for (int k0 = 0; k0 < 512; k0 += 32) {
    __syncthreads();
    *(float4*)&As[la_r][la_c] = *(const float4*)&A[(size_t)(row0 + la_r) * 512 + k0 + la_c];
    *(float4*)&Bs[lb_r][lb_c] = *(const float4*)&Bw[(size_t)(k0 + lb_r) * 512 + col0 + lb_c];
    __syncthreads();
#pragma unroll
    for (int kk = 0; kk < 32; kk += 4) {
      v2f a = *(const v2f*)&As[16 * tr + m][kk + koff];
      v2f b0, b1;
      b0.x = Bs[kk + koff][tc + m];
      b0.y = Bs[kk + koff + 1][tc + m];
      b1.x = Bs[kk + koff][tc + 16 + m];
      b1.y = Bs[kk + koff + 1][tc + 16 + m];
#if __has_builtin(__builtin_amdgcn_wmma_f32_16x16x4_f32)
      acc0 = __builtin_amdgcn_wmma_f32_16x16x4_f32(false, a, false, b0, (short)0, acc0,
                                                   false, false);
      acc1 = __builtin_amdgcn_wmma_f32_16x16x4_f32(false, a, false, b1, (short)0, acc1,
                                                   false, false);
#else
      // compile-survival fallback (wrong math, keeps the build alive)
#pragma unroll
      for (int j = 0; j < 8; ++j) {
        acc0[j] += a.x * b0.x + a.y * b0.y;
        acc1[j] += a.x * b1.x + a.y * b1.y;
      }
#endif
    }
  }

  const int nn    = lane & 15;
  const int mbase = 16 * tr + ((lane < 16) ? 0 : 8);
#pragma unroll
  for (int j = 0; j < 8; ++j) {
    const int mrow = row0 + mbase + j;
    C[(size_t)mrow * 512 + col0 + tc + nn]      = acc0[j] + bias[col0 + tc + nn];
    C[(size_t)mrow * 512 + col0 + tc + 16 + nn] = acc1[j] + bias[col0 + tc + 16 + nn];
  }
}

// agg[b,t,c] = sum_i w[b,i] * Vp[b, (t+idx_i) mod L, c]   (all idx_i < 64)
__global__ void delay_agg(const float* __restrict__ Vp, const int* __restrict__ idx,
                          const float* __restrict__ w, float* __restrict__ agg) {
  __shared__ int   s_idx[TOPK];
  __shared__ float s_w[TOPK];
  const int b  = blockIdx.y;
  const int t0 = blockIdx.x * 32;
  const int c  = threadIdx.x;           // 512
  if (c < TOPK) { s_idx[c] = idx[c]; s_w[c] = w[b * TOPK + c]; }
  __syncthreads();
  const float* Vb = Vp + (size_t)b * LL * DD;
  float* ab       = agg + (size_t)b * LL * DD;
  for (int ts = 0; ts < 32; ++ts) {
    const int t = t0 + ts;
    float acc = 0.f;
#pragma unroll 8
    for (int i = 0; i < TOPK; ++i) {
      const int r = (t + s_idx[i]) & (LL - 1);
      acc += s_w[i] * Vb[(size_t)r * DD + c];
    }
    ab[(size_t)t * DD + c] = acc;
  }
}

extern "C" void kernel_launch(void* const* d_in, const int* in_sizes, int n_in,
                              void* d_out, int out_size, void* d_ws, size_t ws_size,
                              hipStream_t stream) {
  (void)in_sizes; (void)n_in; (void)out_size; (void)ws_size;
  const float* q  = (const float*)d_in[0];
  const float* k  = (const float*)d_in[1];
  const float* v  = (const float*)d_in[2];
  const float* Wq = (const float*)d_in[3];
  const float* bq = (const float*)d_in[4];
  const float* Wk = (const float*)d_in[5];
  const float* bk = (const float*)d_in[6];
  const float* Wv = (const float*)d_in[7];
  const float* bv = (const float*)d_in[8];
  const float* Wo = (const float*)d_in[9];
  const float* bo = (const float*)d_in[10];
  float* out = (float*)d_out;
  float* ws  = (float*)d_ws;

  float* sq  = ws;                 // [B*D]
  float* sk  = ws + 4096;          // [B*D]
  float* mv  = ws + 8192;          // [B*64]
  int*   idx = (int*)(ws + 8704);  // [41]
  float* wts = ws + 8768;          // [B*41]
  float* agg = ws + 16384;         // [B*L*D] = 64 MB
  float* Vp  = out;                // stage V-projection in d_out

  zero_ws<<<64, 256, 0, stream>>>(ws, 16384);
  colsum_qk<<<dim3(32, BB), DD, 0, stream>>>(q, k, sq, sk);
  stats_proj<<<BB, DD, 0, stream>>>(sq, sk, Wq, bq, Wk, bk, mv);
  topk_softmax<<<1, 64, 0, stream>>>(mv, idx, wts);
  gemm512_wmma<<<dim3((BB * LL) / 64, 8), 256, 0, stream>>>(v, Wv, bv, Vp);
  delay_agg<<<dim3(LL / 32, BB), DD, 0, stream>>>(Vp, idx, wts, agg);
  gemm512_wmma<<<dim3((BB * LL) / 64, 8), 256, 0, stream>>>(agg, Wo, bo, out);
}